// RoIPooling2D_14053132992751
// MI455X (gfx1250) — compile-verified
//
#include <hip/hip_runtime.h>
#include <math.h>

// ROI max pooling (Caffe roi_pool), tuned for MI455X (gfx1250, wave32).
//
// Shapes (from reference setup): x=[2,256,50,50] f32, rois=[128,5] f32,
// out=[128,256,7,7] f32. Total traffic ~20MB -> ~1us at 23.3TB/s: this is a
// latency-bound gather/max kernel. Strategy: one workgroup per (roi, 16-ch
// chunk); stage the ROI's feature window into LDS with the CDNA5 async
// global->LDS path (GLOBAL_LOAD_ASYNC_TO_LDS_B32, tracked by ASYNCcnt), then
// compute the 7x7 bin maxes out of LDS. ROI wh <= 256px = 16 cells, so the
// window is <= 18x18 = 324 elems/channel; a >MAXWIN window (impossible for
// this distribution, but kept for safety) falls back to direct global reads.

#define POOLH 7
#define POOLW 7
#define CCH   16      // channels staged per block
#define MAXWIN 324    // 18*18 elems per channel
#define BLOCK 256     // 8 wave32s

typedef __attribute__((address_space(1))) int* gint_p;
typedef __attribute__((address_space(3))) int* lint_p;

__device__ __forceinline__ void async_copy_f32(const float* g, float* l) {
#if defined(__HIP_DEVICE_COMPILE__) && __has_builtin(__builtin_amdgcn_global_load_async_to_lds_b32)
  // gfx1250: each lane copies 4 bytes from the global address g to the LDS address l, tracked with ASYNCcnt.
  __builtin_amdgcn_global_load_async_to_lds_b32(
      (gint_p)g, (lint_p)l, /*imm offset*/0, /*cpol*/0);
#else
  *l = *g;
#endif
}

__device__ __forceinline__ void async_wait_all() {
#if defined(__HIP_DEVICE_COMPILE__) && __has_builtin(__builtin_amdgcn_s_wait_asynccnt)
  __builtin_amdgcn_s_wait_asynccnt(0);
#elif defined(__HIP_DEVICE_COMPILE__) && __has_builtin(__builtin_amdgcn_global_load_async_to_lds_b32)
  asm volatile("s_wait_asynccnt 0" ::: "memory");
#endif
}

__global__ __launch_bounds__(BLOCK) void roipool_kernel(
    const float* __restrict__ x, const float* __restrict__ rois,
    float* __restrict__ out, int C, int H, int W) {
  __shared__ float win[CCH * MAXWIN];

  const int chunks = C / CCH;
  const int n  = blockIdx.x / chunks;
  const int c0 = (blockIdx.x % chunks) * CCH;
  const int tid = threadIdx.x;

  // ---- ROI geometry, bit-exact vs reference (all fp32, RNE rounding) ----
  const float* r = rois + n * 5;
  const int   bidx = (int)r[0];
  const float sx = rintf(r[1] * 0.0625f);   // jnp.round == round-half-even
  const float sy = rintf(r[2] * 0.0625f);
  const float ex = rintf(r[3] * 0.0625f);
  const float ey = rintf(r[4] * 0.0625f);
  const float bsw = fmaxf(ex - sx + 1.0f, 1.0f) / 7.0f;
  const float bsh = fmaxf(ey - sy + 1.0f, 1.0f) / 7.0f;
  const float fH = (float)H, fW = (float)W;

  // Window = [lo(bin0), hi(bin6)) in each dim, computed with the same float
  // ops the reference uses so every bin is a subset of the staged window.
  const int h0 = (int)fminf(fmaxf(sy, 0.0f), fH);
  const int h1 = (int)fminf(fmaxf(ceilf(7.0f * bsh) + sy, 0.0f), fH);
  const int w0 = (int)fminf(fmaxf(sx, 0.0f), fW);
  const int w1 = (int)fminf(fmaxf(ceilf(7.0f * bsw) + sx, 0.0f), fW);
  const int wh = h1 - h0, ww = w1 - w0;
  const int cnt = (wh > 0 && ww > 0) ? wh * ww : 0;
  const bool staged = (cnt > 0) && (cnt <= MAXWIN);

  const size_t plane = (size_t)H * W;
  const float* xb = x + ((size_t)bidx * C + c0) * plane;

  // ---- Stage window for CCH channels: async global -> LDS ----
  if (staged) {
    const int total = CCH * cnt;
    for (int i = tid; i < total; i += BLOCK) {
      const int ch = i / cnt;
      const int j  = i - ch * cnt;
      const int rr = j / ww;
      const int cc = j - rr * ww;
      const float* g = xb + (size_t)ch * plane + (size_t)(h0 + rr) * W + (w0 + cc);
      async_copy_f32(g, &win[i]);
    }
    async_wait_all();
  }
  __syncthreads();

  // ---- Compute 7x7 bin maxes for CCH channels ----
  for (int o = tid; o < CCH * POOLH * POOLW; o += BLOCK) {
    const int ch = o / (POOLH * POOLW);
    const int b  = o - ch * (POOLH * POOLW);
    const int ph = b / POOLW;
    const int pw = b - ph * POOLW;

    const int hs = (int)fminf(fmaxf(floorf((float)ph * bsh) + sy, 0.0f), fH);
    const int he = (int)fminf(fmaxf(ceilf((float)(ph + 1) * bsh) + sy, 0.0f), fH);
    const int ws = (int)fminf(fmaxf(floorf((float)pw * bsw) + sx, 0.0f), fW);
    const int we = (int)fminf(fmaxf(ceilf((float)(pw + 1) * bsw) + sx, 0.0f), fW);

    const bool empty = (he <= hs) || (we <= ws);
    float m = -INFINITY;
    if (!empty) {
      if (staged) {
        const float* wbase = &win[ch * cnt];
        for (int h = hs; h < he; ++h) {
          const float* row = wbase + (h - h0) * ww - w0;
          for (int w = ws; w < we; ++w) m = fmaxf(m, row[w]);
        }
      } else {  // window too large to stage (not reachable for this input dist)
        const float* g = xb + (size_t)ch * plane;
        for (int h = hs; h < he; ++h) {
          const float* row = g + (size_t)h * W;
          for (int w = ws; w < we; ++w) m = fmaxf(m, row[w]);
        }
      }
    }
    // Empty bins produce 0 (reference: where(isfinite, out, 0)).
    out[(((size_t)n * C + (c0 + ch)) * POOLH + ph) * POOLW + pw] = empty ? 0.0f : m;
  }
}

extern "C" void kernel_launch(void* const* d_in, const int* in_sizes, int n_in,
                              void* d_out, int out_size, void* d_ws, size_t ws_size,
                              hipStream_t stream) {
  const float* x    = (const float*)d_in[0];
  const float* rois = (const float*)d_in[1];
  float* out = (float*)d_out;

  const int C = 256, H = 50, W = 50;           // fixed by reference setup
  const int N = in_sizes[1] / 5;               // 128 ROIs
  const int chunks = C / CCH;                  // 16 chunks -> 2048 blocks

  dim3 grid((unsigned)(N * chunks)), block(BLOCK);
  hipLaunchKernelGGL(roipool_kernel, grid, block, 0, stream, x, rois, out, C, H, W);
}